// IMultiHeadAttention2D_47708496724205
// MI455X (gfx1250) — compile-verified
//
#include <hip/hip_runtime.h>
#include <hip/hip_bf16.h>

// ---------------------------------------------------------------------------
// Transposed multi-head attention (B=16, C=256, N=64*64, 8 heads x 64 dim)
// All GEMMs use V_WMMA_F32_16X16X4_F32 (exact fp32; problem is HBM-bound,
// ~77 GFLOP vs ~1.1 GB traffic -> ~50us floor at 23.3 TB/s).
// Each wave computes a 64x16 output column (4 stacked 16x16 WMMA tiles) so the
// streamed B operand is fetched once per 4 WMMAs instead of once per WMMA.
// ---------------------------------------------------------------------------

typedef __attribute__((ext_vector_type(2))) float v2f;
typedef __attribute__((ext_vector_type(8))) float v8f;

#define NUM_HEADS 8
#define HEAD_DIM  64
#define EMBED     512      // NUM_HEADS*HEAD_DIM
#define BATCH     16
#define CIN       256
#define NTOK      4096     // 64*64
#define QKV_ROWS  1536     // 3*EMBED
#define SCALE     0.125f   // 1/sqrt(64)

__device__ __forceinline__ v8f wmma16x16x4(v2f a, v2f b, v8f c) {
  // D = A(16x4, f32) * B(4x16, f32) + C(16x16, f32)
  return __builtin_amdgcn_wmma_f32_16x16x4_f32(false, a, false, b, (short)0, c,
                                               false, false);
}

// One wave computes a 64(M) x 16(N) fp32 column: 4 stacked 16x16 tiles.
// A is 64 x K (row-major, lda), B is K x >=16 (row-major, ldb).
// Fragment layout per CDNA5 ISA 7.12.2 (wave32):
//   A: lane (m = lane&15) holds K = {0,1} (lanes 0-15) / {2,3} (lanes 16-31)
//   B: lane (n = lane&15) holds K = {0,1} (lanes 0-15) / {2,3} (lanes 16-31)
// The B fragment is loaded once per k-step and reused by all 4 WMMAs.
__device__ __forceinline__ void gemm_col_64x16(const float* __restrict__ A, int lda,
                                               const float* __restrict__ B, int ldb,
                                               int K, int lane, v8f acc[4]) {
  const int l16 = lane & 15;
  const int khi = (lane >> 4) * 2;  // 0 or 2
  const float* a0   = A + (size_t)l16 * lda + khi;
  const float* bcol = B + (size_t)khi * ldb + l16;
#pragma unroll 4
  for (int k = 0; k < K; k += 4) {
    v2f b = { bcol[(size_t)k * ldb], bcol[(size_t)(k + 1) * ldb] };
#pragma unroll
    for (int t = 0; t < 4; ++t) {
      const float* ar = a0 + (size_t)(t * 16) * lda;
      v2f a = { ar[k], ar[k + 1] };
      acc[t] = wmma16x16x4(a, b, acc[t]);
    }
  }
}

// ---------------------------------------------------------------------------
// Kernel 1: qkv[b, e, n] = sum_c w_qkv[e, c] * x[b, c, n]
//   per batch: (1536 x 256) * (256 x 4096)
//   block = 8 waves -> 64(M) x 128(N) tile; grid = B * 24 * 32
// ---------------------------------------------------------------------------
__global__ void __launch_bounds__(256) qkv_gemm_kernel(
    const float* __restrict__ x, const float* __restrict__ w_qkv,
    float* __restrict__ qkv) {
  const int ntn = NTOK / 128;      // 32
  const int ntm = QKV_ROWS / 64;   // 24
  int bid = blockIdx.x;
  int b   = bid / (ntm * ntn);
  int r   = bid % (ntm * ntn);
  int tm  = r / ntn;
  int tn  = r % ntn;

  const int wave = threadIdx.x >> 5;
  const int lane = threadIdx.x & 31;
  const int m0 = tm * 64;
  const int n0 = tn * 128 + wave * 16;

  const float* A = w_qkv + (size_t)m0 * CIN;             // 64 x 256
  const float* B = x + (size_t)b * CIN * NTOK + n0;      // 256 x 4096 strip
  v8f acc[4] = {};
  gemm_col_64x16(A, CIN, B, NTOK, CIN, lane, acc);

  float* C = qkv + (size_t)b * QKV_ROWS * NTOK + (size_t)m0 * NTOK + n0;
  const int l16 = lane & 15, half = lane >> 4;
#pragma unroll
  for (int t = 0; t < 4; ++t)
#pragma unroll
    for (int rr = 0; rr < 8; ++rr)
      C[(size_t)(t * 16 + 8 * half + rr) * NTOK + l16] = acc[t][rr];
}

// ---------------------------------------------------------------------------
// Kernel 2: one workgroup per (b, h).
//   att(64x64) = SCALE * q(64x4096) * k^T ; softmax rows ; out = att * v
// ---------------------------------------------------------------------------
__global__ void __launch_bounds__(256) attn_kernel(
    const float* __restrict__ qkv, float* __restrict__ aout) {
  const int b = blockIdx.x / NUM_HEADS;
  const int h = blockIdx.x % NUM_HEADS;
  const size_t base = (size_t)b * QKV_ROWS * NTOK;
  const float* q = qkv + base + (size_t)(0 * EMBED + h * HEAD_DIM) * NTOK;
  const float* k = qkv + base + (size_t)(1 * EMBED + h * HEAD_DIM) * NTOK;
  const float* v = qkv + base + (size_t)(2 * EMBED + h * HEAD_DIM) * NTOK;

  __shared__ float att[64 * 64];   // 16 KB of the WGP's 320 KB LDS

  const int wave = threadIdx.x >> 5;
  const int lane = threadIdx.x & 31;
  const int l16 = lane & 15, half = lane >> 4;
  const int khi = half * 2;

  // ---- Stage 1: att = q * k^T (M=N=64, K=4096). Wave w -> row-tile w>>1,
  //      two adjacent col-tiles; the q (A) fragment is shared by both WMMAs.
  {
    const int tm  = wave >> 1;
    const int tn0 = (wave & 1) * 2;
    const float* arow = q + (size_t)(tm * 16 + l16) * NTOK + khi;
    // B[s][j] = k[j][s] -> per-lane access mirrors the A fragment
    const float* br0 = k + (size_t)(tn0 * 16 + l16) * NTOK + khi;
    const float* br1 = br0 + (size_t)16 * NTOK;
    v8f acc0 = {}, acc1 = {};
#pragma unroll 4
    for (int kk = 0; kk < NTOK; kk += 4) {
      v2f a  = { arow[kk], arow[kk + 1] };
      v2f b0 = { br0[kk],  br0[kk + 1] };
      v2f b1 = { br1[kk],  br1[kk + 1] };
      acc0 = wmma16x16x4(a, b0, acc0);
      acc1 = wmma16x16x4(a, b1, acc1);
    }
#pragma unroll
    for (int rr = 0; rr < 8; ++rr) {
      const int row = tm * 16 + 8 * half + rr;
      att[row * 64 + tn0 * 16 + l16]      = acc0[rr] * SCALE;
      att[row * 64 + (tn0 + 1) * 16 + l16] = acc1[rr] * SCALE;
    }
  }
  __syncthreads();

  // ---- Stage 2: row softmax (64 rows, one thread each) ----
  if (threadIdx.x < 64) {
    float* row = &att[threadIdx.x * 64];
    float m = row[0];
#pragma unroll 8
    for (int j = 1; j < 64; ++j) m = fmaxf(m, row[j]);
    float s = 0.f;
#pragma unroll 8
    for (int j = 0; j < 64; ++j) {
      float e = __expf(row[j] - m);
      row[j] = e;
      s += e;
    }
    const float inv = 1.f / s;
#pragma unroll 8
    for (int j = 0; j < 64; ++j) row[j] *= inv;
  }
  __syncthreads();

  // ---- Stage 3: out = att(64x64) * v(64x4096). Each wave takes one 16-col
  //      strip of v at a time and produces the full 64-row column; the v (B)
  //      fragment is fetched once per k-step for all 4 WMMAs.
  float* obase = aout + (size_t)b * EMBED * NTOK + (size_t)h * HEAD_DIM * NTOK;
  for (int tn = wave; tn < NTOK / 16; tn += 8) {
    const float* bcol = v + (size_t)khi * NTOK + tn * 16 + l16;
    v8f acc[4] = {};
#pragma unroll 4
    for (int kk = 0; kk < 64; kk += 4) {
      v2f bb = { bcol[(size_t)kk * NTOK], bcol[(size_t)(kk + 1) * NTOK] };
#pragma unroll
      for (int t = 0; t < 4; ++t) {
        const float* ar = &att[(t * 16 + l16) * 64 + khi];
        v2f a = { ar[kk], ar[kk + 1] };
        acc[t] = wmma16x16x4(a, bb, acc[t]);
      }
    }
#pragma unroll
    for (int t = 0; t < 4; ++t)
#pragma unroll
      for (int rr = 0; rr < 8; ++rr)
        obase[(size_t)(t * 16 + 8 * half + rr) * NTOK + tn * 16 + l16] =
            acc[t][rr];
  }
}

// ---------------------------------------------------------------------------
// Kernel 3: out[b, c, n] = sum_e w_out[c, e] * aout[b, e, n] + b_out[c]
//   per batch: (256 x 512) * (512 x 4096); block = 64(M) x 128(N)
// ---------------------------------------------------------------------------
__global__ void __launch_bounds__(256) out_gemm_kernel(
    const float* __restrict__ aout, const float* __restrict__ w_out,
    const float* __restrict__ b_out, float* __restrict__ out) {
  const int ntn = NTOK / 128;  // 32
  const int ntm = CIN / 64;    // 4
  int bid = blockIdx.x;
  int b   = bid / (ntm * ntn);
  int r   = bid % (ntm * ntn);
  int tm  = r / ntn;
  int tn  = r % ntn;

  const int wave = threadIdx.x >> 5;
  const int lane = threadIdx.x & 31;
  const int m0 = tm * 64;
  const int n0 = tn * 128 + wave * 16;

  const float* A = w_out + (size_t)m0 * EMBED;             // 64 x 512
  const float* B = aout + (size_t)b * EMBED * NTOK + n0;   // 512 x 4096 strip
  v8f acc[4] = {};
  gemm_col_64x16(A, EMBED, B, NTOK, EMBED, lane, acc);

  float* C = out + (size_t)b * CIN * NTOK + (size_t)m0 * NTOK + n0;
  const int l16 = lane & 15, half = lane >> 4;
#pragma unroll
  for (int t = 0; t < 4; ++t)
#pragma unroll
    for (int rr = 0; rr < 8; ++rr) {
      const int row = t * 16 + 8 * half + rr;
      C[(size_t)row * NTOK + l16] = acc[t][rr] + b_out[m0 + row];
    }
}

// ---------------------------------------------------------------------------
extern "C" void kernel_launch(void* const* d_in, const int* in_sizes, int n_in,
                              void* d_out, int out_size, void* d_ws, size_t ws_size,
                              hipStream_t stream) {
  const float* x     = (const float*)d_in[0];  // (16, 256, 64, 64)
  const float* w_qkv = (const float*)d_in[1];  // (1536, 256)
  const float* w_out = (const float*)d_in[2];  // (256, 512)
  const float* b_out = (const float*)d_in[3];  // (256,)
  float* out = (float*)d_out;                  // (16, 256, 64, 64)

  // Scratch: qkv (16*1536*4096 f32 = 403 MB) + attn-out (16*512*4096 = 134 MB)
  float* qkv  = (float*)d_ws;
  float* aout = qkv + (size_t)BATCH * QKV_ROWS * NTOK;

  qkv_gemm_kernel<<<dim3(BATCH * (QKV_ROWS / 64) * (NTOK / 128)), dim3(256), 0,
                    stream>>>(x, w_qkv, qkv);
  attn_kernel<<<dim3(BATCH * NUM_HEADS), dim3(256), 0, stream>>>(qkv, aout);
  out_gemm_kernel<<<dim3(BATCH * (CIN / 64) * (NTOK / 128)), dim3(256), 0,
                    stream>>>(aout, w_out, b_out, out);
}